// ContextRGAT_43233140801937
// MI455X (gfx1250) — compile-verified
//
#include <hip/hip_runtime.h>

// ---------------------------------------------------------------------------
// RGAT (2x RGATConv, heads=1) for MI455X / gfx1250.
// N=50000 nodes, E=500000 edges, R=3 relations, C=128 channels.
//
// Design (see analysis): GEMM part is ~10 GFLOP total -> negligible; keep it
// fp32 via V_WMMA_F32_16X16X4_F32. Edge gather/scatter dominates; xw (76.8MB)
// fits in the 192MB L2 so random row gathers + f32 atomics stay on-chip.
// ---------------------------------------------------------------------------

typedef float v2f __attribute__((ext_vector_type(2)));
typedef float v8f __attribute__((ext_vector_type(8)));

#define CDIM 128

// Order-preserving float <-> uint key (for atomicMax on signed floats).
__device__ __forceinline__ unsigned f2key(float f) {
  unsigned b = __float_as_uint(f);
  return (b & 0x80000000u) ? ~b : (b | 0x80000000u);
}
__device__ __forceinline__ float key2f(unsigned u) {
  unsigned b = (u & 0x80000000u) ? (u & 0x7FFFFFFFu) : ~u;
  return __uint_as_float(b);
}

__global__ void rgat_zero_u32(unsigned* __restrict__ p, size_t n) {
  size_t i = (size_t)blockIdx.x * blockDim.x + threadIdx.x;
  if (i < n) p[i] = 0u;
}

// ---------------------------------------------------------------------------
// xw[r] = X @ W[r], fp32 WMMA. Block = 128 threads = 4 waves; each wave owns
// one 16x16 output tile (same 16 rows, 4 consecutive 16-col tiles -> A rows
// reused out of L0). K advanced 4 at a time with V_WMMA_F32_16X16X4_F32.
// A layout (16x4 f32): lanes 0-15 hold M=lane, v0=K0 v1=K1; lanes 16-31 K2/K3.
// B layout (4x16 f32): lanes 0-15 hold N=lane, v0=K0 v1=K1; lanes 16-31 K2/K3.
// C/D layout (16x16 f32): N = lane&15, VGPR v -> M = v + 8*(lane>>4).
// ---------------------------------------------------------------------------
__global__ __launch_bounds__(128) void rgat_gemm_wmma_f32(
    const float* __restrict__ X, const float* __restrict__ W,
    float* __restrict__ XW, int nRows) {
  const int lane = threadIdx.x & 31;
  const int wv   = threadIdx.x >> 5;
  const int m0   = blockIdx.x << 4;
  const int n0   = (blockIdx.y << 6) + (wv << 4);
  const int r    = blockIdx.z;

  const float* Wr  = W + (size_t)r * CDIM * CDIM;
  float*       Out = XW + (size_t)r * (size_t)nRows * CDIM;

  const int li = lane & 15;
  const int hi = lane >> 4;

  const float* arow = X  + (size_t)(m0 + li) * CDIM + (hi << 1);
  const float* bcol = Wr + (size_t)(hi << 1) * CDIM + n0 + li;

  v8f c = {};
#pragma unroll 8
  for (int k = 0; k < CDIM; k += 4) {
    v2f a, b;
    a.x = arow[k];
    a.y = arow[k + 1];
    b.x = bcol[(size_t)k * CDIM];
    b.y = bcol[(size_t)(k + 1) * CDIM];
    c = __builtin_amdgcn_wmma_f32_16x16x4_f32(
        /*neg_a=*/false, a, /*neg_b=*/false, b,
        /*c_mod=*/(short)0, c, /*reuse_a=*/false, /*reuse_b=*/false);
  }

#pragma unroll
  for (int v = 0; v < 8; ++v)
    Out[(size_t)(m0 + v + (hi << 3)) * CDIM + n0 + li] = c[v];
}

// sq[i] = xw_row[i] . q ; sk[i] = xw_row[i] . k   for i in [0, R*N)
__global__ void rgat_sqk(const float* __restrict__ xw,
                         const float* __restrict__ q,
                         const float* __restrict__ k,
                         float* __restrict__ sq, float* __restrict__ sk,
                         int total) {
  int i = blockIdx.x * blockDim.x + threadIdx.x;
  if (i >= total) return;
  const float4* row = (const float4*)(xw + (size_t)i * CDIM);
  const float4* q4  = (const float4*)q;
  const float4* k4  = (const float4*)k;
  float aq = 0.f, ak = 0.f;
#pragma unroll 8
  for (int j = 0; j < CDIM / 4; ++j) {
    float4 v = row[j], qq = q4[j], kk = k4[j];
    aq += v.x * qq.x + v.y * qq.y + v.z * qq.z + v.w * qq.w;
    ak += v.x * kk.x + v.y * kk.y + v.z * kk.z + v.w * kk.w;
  }
  sq[i] = aq;
  sk[i] = ak;
}

// scal = dot(le, e) over 128 elements (edge_attr has dim 1 -> scalar path)
__global__ void rgat_dot128(const float* __restrict__ le,
                            const float* __restrict__ e,
                            float* __restrict__ out) {
  __shared__ float s[CDIM];
  int t = threadIdx.x;
  s[t] = le[t] * e[t];
  __syncthreads();
  for (int off = CDIM / 2; off > 0; off >>= 1) {
    if (t < off) s[t] += s[t + off];
    __syncthreads();
  }
  if (t == 0) out[0] = s[0];
}

// Pass 1: logits + leaky-relu, segment max via keyed atomicMax.
__global__ void rgat_alpha(const int* __restrict__ src, const int* __restrict__ dst,
                           const int* __restrict__ et, const float* __restrict__ ea,
                           const float* __restrict__ sq, const float* __restrict__ sk,
                           const float* __restrict__ scal,
                           float* __restrict__ alpha, unsigned* __restrict__ amax,
                           int E, int N) {
  int e = blockIdx.x * blockDim.x + threadIdx.x;
  if (e >= E) return;
  int s = src[e], d = dst[e], r = et[e];
  float a = sq[(size_t)r * N + d] + sk[(size_t)r * N + s] + ea[e] * scal[0];
  a = (a > 0.f) ? a : 0.2f * a;  // leaky_relu, slope 0.2
  alpha[e] = a;
  atomicMax(&amax[d], f2key(a));
}

// Pass 2: ex = exp(a - amax[dst]); denom[dst] += ex. (alpha buf overwritten)
__global__ void rgat_expsum(const int* __restrict__ dst,
                            float* __restrict__ alpha,
                            const unsigned* __restrict__ amax,
                            float* __restrict__ denom, int E) {
  int e = blockIdx.x * blockDim.x + threadIdx.x;
  if (e >= E) return;
  int d = dst[e];
  float m = key2f(amax[d]);
  if ((__float_as_uint(m) & 0x7F800000u) == 0x7F800000u) m = 0.f;  // !isfinite
  float ex = expf(alpha[e] - m);
  alpha[e] = ex;
  atomicAdd(&denom[d], ex);
}

// Pass 3: out[dst,:] += (ex/denom[dst]) * xw[et, src, :]. One 32-lane wave per
// edge; each lane moves one float4 (512B row -> 4 f32 atomics per lane).
__global__ void rgat_agg(const int* __restrict__ src, const int* __restrict__ dst,
                         const int* __restrict__ et,
                         const float* __restrict__ alpha,
                         const float* __restrict__ denom,
                         const float* __restrict__ xw,
                         float* __restrict__ out, int E, int N) {
  unsigned t = blockIdx.x * blockDim.x + threadIdx.x;
  int e = (int)(t >> 5);
  if (e >= E) return;
  int lane = (int)(t & 31u);
  int d = dst[e];
  float coef = alpha[e] / (denom[d] + 1e-16f);
  const float4 v =
      *(const float4*)(xw + ((size_t)et[e] * N + src[e]) * CDIM + lane * 4);
  float* o = out + (size_t)d * CDIM + lane * 4;
  atomicAdd(o + 0, coef * v.x);
  atomicAdd(o + 1, coef * v.y);
  atomicAdd(o + 2, coef * v.z);
  atomicAdd(o + 3, coef * v.w);
}

// Fused bias (+optional ReLU) in place.
__global__ void rgat_bias_act(float* __restrict__ o, const float* __restrict__ b,
                              size_t total, int do_relu) {
  size_t i = (size_t)blockIdx.x * blockDim.x + threadIdx.x;
  if (i >= total) return;
  float v = o[i] + b[i & (CDIM - 1)];
  o[i] = (do_relu && v < 0.f) ? 0.f : v;
}

// ---------------------------------------------------------------------------
extern "C" void kernel_launch(void* const* d_in, const int* in_sizes, int n_in,
                              void* d_out, int out_size, void* d_ws, size_t ws_size,
                              hipStream_t stream) {
  const float* x   = (const float*)d_in[0];
  const int*   ei  = (const int*)d_in[1];
  const int*   et  = (const int*)d_in[2];
  const float* ea  = (const float*)d_in[3];
  const float* w1  = (const float*)d_in[4];
  const float* q1  = (const float*)d_in[5];
  const float* k1  = (const float*)d_in[6];
  const float* le1 = (const float*)d_in[7];
  const float* e1  = (const float*)d_in[8];
  const float* b1  = (const float*)d_in[9];
  const float* w2  = (const float*)d_in[10];
  const float* q2  = (const float*)d_in[11];
  const float* k2  = (const float*)d_in[12];
  const float* le2 = (const float*)d_in[13];
  const float* e2  = (const float*)d_in[14];
  const float* b2  = (const float*)d_in[15];

  const int C = CDIM;
  const int N = in_sizes[0] / C;          // 50000
  const int E = in_sizes[2];              // 500000
  const int R = in_sizes[4] / (C * C);    // 3
  const int* src = ei;
  const int* dst = ei + E;

  // Workspace layout (floats): xw | h | sq | sk | denom | amax | alpha | scal
  float*    xw     = (float*)d_ws;                       // R*N*C  (76.8 MB)
  float*    h      = xw + (size_t)R * N * C;             // N*C    (25.6 MB)
  float*    sq     = h + (size_t)N * C;                  // R*N
  float*    sk     = sq + (size_t)R * N;                 // R*N
  float*    denom  = sk + (size_t)R * N;                 // N
  unsigned* amax   = (unsigned*)(denom + N);             // N
  float*    alphab = (float*)(amax + N);                 // E
  float*    scal   = alphab + E;                         // 1

  auto layer = [&](const float* in, const float* W, const float* q,
                   const float* k, const float* le, const float* ev,
                   const float* b, float* out, int relu) {
    dim3 gg(N / 16, C / 64, R);
    rgat_gemm_wmma_f32<<<gg, 128, 0, stream>>>(in, W, xw, N);

    int rn = R * N;
    rgat_sqk<<<(rn + 255) / 256, 256, 0, stream>>>(xw, q, k, sq, sk, rn);
    rgat_dot128<<<1, CDIM, 0, stream>>>(le, ev, scal);

    size_t zn = (size_t)N * C;
    rgat_zero_u32<<<(unsigned)((zn + 255) / 256), 256, 0, stream>>>(
        (unsigned*)out, zn);
    rgat_zero_u32<<<(N + 255) / 256, 256, 0, stream>>>(amax, (size_t)N);
    rgat_zero_u32<<<(N + 255) / 256, 256, 0, stream>>>((unsigned*)denom,
                                                       (size_t)N);

    rgat_alpha<<<(E + 255) / 256, 256, 0, stream>>>(src, dst, et, ea, sq, sk,
                                                    scal, alphab, amax, E, N);
    rgat_expsum<<<(E + 255) / 256, 256, 0, stream>>>(dst, alphab, amax, denom, E);

    size_t tn = (size_t)E * 32;
    rgat_agg<<<(unsigned)((tn + 255) / 256), 256, 0, stream>>>(
        src, dst, et, alphab, denom, xw, out, E, N);

    rgat_bias_act<<<(unsigned)((zn + 255) / 256), 256, 0, stream>>>(out, b, zn,
                                                                    relu);
  };

  layer(x, w1, q1, k1, le1, e1, b1, h, 1);
  layer(h, w2, q2, k2, le2, e2, b2, (float*)d_out, 0);
}